// KNNDownSampling_45827301048343
// MI455X (gfx1250) — compile-verified
//
#include <hip/hip_runtime.h>

typedef __attribute__((ext_vector_type(2))) float v2f;
typedef __attribute__((ext_vector_type(8))) float v8f;
typedef unsigned int u32x4_t __attribute__((ext_vector_type(4)));
typedef int i32x4_t __attribute__((ext_vector_type(4)));
typedef int i32x8_t __attribute__((ext_vector_type(8)));

#define B_   16
#define S_   4096
#define F_   256
#define K_   512
#define P_   8
#define ITS_ 5

// ---------------- TDM: 1-D global->LDS DMA copy of `ndwords` dwords ----------------
// D# group0: count=1 (valid, user mode), lds_addr, 57-bit global_addr, type=2.
// D# group1: data_size=4B, tensor_dim0 = tile_dim0 = ndwords (1 row), dim1=1,
//            tensor_dim0_stride = ndwords, no padding / iteration / multicast.
// Groups 2/3 zero (tensor <= 2D). Tracked by TENSORcnt.
__device__ __forceinline__ unsigned lds_off_of(const void* p) {
    // generic LDS pointer: low 32 bits are the LDS byte offset (aperture rule)
    return (unsigned)(uintptr_t)p;
}

__device__ __forceinline__ void tdm_copy_dwords_to_lds(const void* gsrc,
                                                       unsigned lds_byte_off,
                                                       unsigned ndwords) {
    unsigned long long ga = (unsigned long long)(uintptr_t)gsrc;
    u32x4_t g0;
    g0[0] = 1u;                                              // count=1, user descriptor
    g0[1] = lds_byte_off;                                    // lds_addr [63:32]
    g0[2] = (unsigned)ga;                                    // global_addr[31:0] -> bits [95:64]
    g0[3] = (unsigned)((ga >> 32) & 0x01ffffffu)             // global_addr[56:32]
          | 0x80000000u;                                     // type=2 in bits [127:126]
    i32x8_t g1;
    g1[0] = (int)(2u << 16);                                 // workgroup_mask=0, data_size=2 (4B)
    g1[1] = (int)((ndwords & 0xffffu) << 16);                // tensor_dim0[15:0] in bits [63:48]
    g1[2] = (int)(((ndwords >> 16) & 0xffffu) | (1u << 16)); // tensor_dim0[31:16], tensor_dim1=1
    g1[3] = (int)((ndwords & 0xffffu) << 16);                // tile_dim0 in bits [127:112]
    g1[4] = 0;                                               // tile_dim1=0, tile_dim2=0 (unused)
    g1[5] = (int)ndwords;                                    // tensor_dim0_stride[31:0]
    g1[6] = 0;
    g1[7] = 0;
    i32x4_t z4 = {0, 0, 0, 0};
#if __clang_major__ >= 23
    i32x8_t z8 = {0, 0, 0, 0, 0, 0, 0, 0};
    __builtin_amdgcn_tensor_load_to_lds(g0, g1, z4, z4, z8, 0);
#else
    __builtin_amdgcn_tensor_load_to_lds(g0, g1, z4, z4, 0);
#endif
}

// ---------------- kernel: init centroids c = positions[:, :K, :] ----------------
__global__ void k_init_centroids(const float* __restrict__ pos, float* __restrict__ c) {
    int i = blockIdx.x * blockDim.x + threadIdx.x;   // over B*K*3
    if (i >= B_ * K_ * 3) return;
    int b = i / (K_ * 3);
    int r = i % (K_ * 3);
    c[i] = pos[(size_t)b * S_ * 3 + r];
}

// ---------------- kernel: k-means assignment via V_WMMA_F32_16X16X4_F32 ----------------
// Each wave owns a 16-point tile (M). Loops over 32 centroid tiles (N) of 16.
// A row (point m)     = [px, py, pz, 1]
// B col (centroid n)  = [-2cx, -2cy, -2cz, |c|^2]
// D[m][n] = -2 p.c + |c|^2  -> argmin over n == nearest centroid.
__global__ void k_assign(const float* __restrict__ pos, const float* __restrict__ c,
                         int* __restrict__ assign) {
    __shared__ float  craw[K_ * 3];   // raw centroids via TDM (6 KB)
    __shared__ float4 enc[K_];        // encoded B columns (8 KB)
    int b = blockIdx.x >> 5;          // 32 blocks per batch

    if (threadIdx.x < 32) {           // wave 0 issues the DMA, waits on TENSORcnt
        tdm_copy_dwords_to_lds(c + (size_t)b * K_ * 3, lds_off_of(craw), K_ * 3);
        __builtin_amdgcn_s_wait_tensorcnt(0);
    }
    __syncthreads();
    for (int j = threadIdx.x; j < K_; j += 256) {
        float cx = craw[3 * j], cy = craw[3 * j + 1], cz = craw[3 * j + 2];
        enc[j] = make_float4(-2.f * cx, -2.f * cy, -2.f * cz, cx * cx + cy * cy + cz * cz);
    }
    __syncthreads();

    int wave = threadIdx.x >> 5;
    int lane = threadIdx.x & 31;
    int tile = (blockIdx.x & 31) * 8 + wave;  // 0..255 tile within batch
    int s0   = tile * 16;
    int m    = lane & 15;
    int kh   = lane >> 4;                     // lane half selects K pair {0,1} / {2,3}

    // A operand: 16x4 f32, VGPR0 = A[m][2*kh], VGPR1 = A[m][2*kh+1]
    const float* p = pos + ((size_t)b * S_ + s0 + m) * 3;
    float pc[4] = { p[0], p[1], p[2], 1.0f };
    v2f a; a.x = pc[2 * kh]; a.y = pc[2 * kh + 1];

    float best[8];
    int   bidx[8];
#pragma unroll
    for (int r = 0; r < 8; ++r) { best[r] = 3.4e38f; bidx[r] = 0x7fffffff; }

    int n = lane & 15;
    for (int ct = 0; ct < K_ / 16; ++ct) {
        float4 e = enc[ct * 16 + n];
        float ec[4] = { e.x, e.y, e.z, e.w };
        v2f bb; bb.x = ec[2 * kh]; bb.y = ec[2 * kh + 1];
        v8f acc = {};
        acc = __builtin_amdgcn_wmma_f32_16x16x4_f32(
            /*neg_a=*/false, a, /*neg_b=*/false, bb,
            /*c_mod=*/(short)0, acc, /*reuse_a=*/false, /*reuse_b=*/false);
        int j = ct * 16 + n;
#pragma unroll
        for (int r = 0; r < 8; ++r) {
            float v = acc[r];
            if (v < best[r] || (v == best[r] && j < bidx[r])) { best[r] = v; bidx[r] = j; }
        }
    }
    // lex-(value,index) argmin across the 16 lanes of each half (masks stay in-half)
#pragma unroll
    for (int mask = 1; mask < 16; mask <<= 1) {
#pragma unroll
        for (int r = 0; r < 8; ++r) {
            float ov = __shfl_xor(best[r], mask, 32);
            int   oi = __shfl_xor(bidx[r], mask, 32);
            if (ov < best[r] || (ov == best[r] && oi < bidx[r])) { best[r] = ov; bidx[r] = oi; }
        }
    }
    // D layout: VGPR r -> M = r + 8*half
    if ((lane & 15) == 0) {
#pragma unroll
        for (int r = 0; r < 8; ++r)
            assign[(size_t)b * S_ + s0 + r + 8 * kh] = bidx[r];
    }
}

// ---------------- kernel: centroid update (deterministic, no atomics) ----------------
__global__ void k_update(const float* __restrict__ pos, const int* __restrict__ assign,
                         float* __restrict__ c) {
    __shared__ float sp[S_ * 3];    // 48 KB positions via TDM
    __shared__ int   sa[S_];        // 16 KB assignments via TDM
    int b    = blockIdx.x >> 1;     // 2 blocks per batch
    int half = blockIdx.x & 1;

    if (threadIdx.x < 32) {
        tdm_copy_dwords_to_lds(pos + (size_t)b * S_ * 3, lds_off_of(sp), S_ * 3);
        tdm_copy_dwords_to_lds(assign + (size_t)b * S_, lds_off_of(sa), S_);
        __builtin_amdgcn_s_wait_tensorcnt(0);
    }
    __syncthreads();

    int wave = threadIdx.x >> 5, lane = threadIdx.x & 31;
    for (int i = 0; i < 32; ++i) {
        int k = half * 256 + wave * 32 + i;
        float cnt = 0.f, ax = 0.f, ay = 0.f, az = 0.f;
        for (int s = lane; s < S_; s += 32) {
            if (sa[s] == k) {
                cnt += 1.f;
                ax += sp[3 * s]; ay += sp[3 * s + 1]; az += sp[3 * s + 2];
            }
        }
#pragma unroll
        for (int mask = 16; mask >= 1; mask >>= 1) {
            cnt += __shfl_xor(cnt, mask, 32);
            ax  += __shfl_xor(ax,  mask, 32);
            ay  += __shfl_xor(ay,  mask, 32);
            az  += __shfl_xor(az,  mask, 32);
        }
        if (lane == 0) {
            float d = fmaxf(cnt, 1.f);
            float* cp = c + ((size_t)b * K_ + k) * 3;
            cp[0] = ax / d; cp[1] = ay / d; cp[2] = az / d;
        }
    }
}

// ---------------- kernel: per-centroid top-8 nearest points (stable order) ----------------
// key = |p|^2 - 2 c.p  (|c|^2 constant per centroid -> dropped; preserves order).
// P successive lex-min passes reproduce jax.lax.top_k tie-breaking (lowest index first).
__global__ void k_topk(const float* __restrict__ pos, const float* __restrict__ c,
                       int* __restrict__ idxout) {
    __shared__ float sp[S_ * 3];    // 48 KB positions via TDM
    __shared__ float sn[S_];        // 16 KB |p|^2 computed in LDS
    int b = blockIdx.x >> 2;        // 4 blocks per batch
    int q = blockIdx.x & 3;

    if (threadIdx.x < 32) {
        tdm_copy_dwords_to_lds(pos + (size_t)b * S_ * 3, lds_off_of(sp), S_ * 3);
        __builtin_amdgcn_s_wait_tensorcnt(0);
    }
    __syncthreads();
    for (int s = threadIdx.x; s < S_; s += 256) {
        float x = sp[3 * s], y = sp[3 * s + 1], z = sp[3 * s + 2];
        sn[s] = x * x + y * y + z * z;
    }
    __syncthreads();

    int wave = threadIdx.x >> 5, lane = threadIdx.x & 31;
    for (int i = 0; i < 16; ++i) {
        int k = q * 128 + wave * 16 + i;
        const float* cp = c + ((size_t)b * K_ + k) * 3;
        float cx = cp[0], cy = cp[1], cz = cp[2];
        float pkey = -3.4e38f; int pidx = -1;
        for (int pass = 0; pass < P_; ++pass) {
            float bk = 3.4e38f; int bi = 0x7fffffff;
            for (int s = lane; s < S_; s += 32) {
                float key = sn[s] - 2.f * (cx * sp[3 * s] + cy * sp[3 * s + 1] + cz * sp[3 * s + 2]);
                bool gtprev = (key > pkey) || (key == pkey && s > pidx);
                if (gtprev && (key < bk || (key == bk && s < bi))) { bk = key; bi = s; }
            }
#pragma unroll
            for (int mask = 1; mask < 32; mask <<= 1) {
                float ov = __shfl_xor(bk, mask, 32);
                int   oi = __shfl_xor(bi, mask, 32);
                if (ov < bk || (ov == bk && oi < bi)) { bk = ov; bi = oi; }
            }
            pkey = bk; pidx = bi;
            if (lane == 0) idxout[((size_t)b * K_ + k) * P_ + pass] = bi;
        }
    }
}

// ---------------- kernel: gather 8 feature rows per centroid and average ----------------
__global__ void k_gather(const float* __restrict__ x, const int* __restrict__ idx,
                         float* __restrict__ out) {
    int bk = blockIdx.x;            // 0..B*K-1
    int f  = threadIdx.x;           // 0..255 (F)
    int b  = bk >> 9;               // /K
    const float* xb = x + (size_t)b * S_ * F_;
    int js[P_];
#pragma unroll
    for (int p = 0; p < P_; ++p) {
        js[p] = idx[(size_t)bk * P_ + p];
        __builtin_prefetch(&xb[(size_t)js[p] * F_ + f], 0, 1);   // global_prefetch_b8
    }
    float s = 0.f;
#pragma unroll
    for (int p = 0; p < P_; ++p)
        s += xb[(size_t)js[p] * F_ + f];
    out[(size_t)bk * F_ + f] = s * 0.125f;
}

extern "C" void kernel_launch(void* const* d_in, const int* in_sizes, int n_in,
                              void* d_out, int out_size, void* d_ws, size_t ws_size,
                              hipStream_t stream) {
    (void)in_sizes; (void)n_in; (void)out_size; (void)ws_size;
    const float* x   = (const float*)d_in[0];
    const float* pos = (const float*)d_in[1];
    float* out = (float*)d_out;

    char* ws = (char*)d_ws;
    float* c    = (float*)ws;                                  // B*K*3 f32  = 98304 B
    int* assign = (int*)(ws + 98304);                          // B*S  i32   = 262144 B
    int* idx    = (int*)(ws + 98304 + 262144);                 // B*K*P i32  = 262144 B

    k_init_centroids<<<(B_ * K_ * 3 + 255) / 256, 256, 0, stream>>>(pos, c);
    for (int it = 0; it < ITS_; ++it) {
        k_assign<<<512, 256, 0, stream>>>(pos, c, assign);     // WMMA distance tiles
        k_update<<<32, 256, 0, stream>>>(pos, assign, c);
    }
    k_topk<<<64, 256, 0, stream>>>(pos, c, idx);
    k_gather<<<B_ * K_, 256, 0, stream>>>(x, idx, out);
}